// MetaKNetwork_21534966022155
// MI455X (gfx1250) — compile-verified
//
#include <hip/hip_runtime.h>
#include <math.h>

typedef __attribute__((ext_vector_type(2))) float v2f;
typedef __attribute__((ext_vector_type(8))) float v8f;

#define KNN 32
#define ROWS_PER_WAVE 16
#define WAVES_PER_BLOCK 8
#define ROWS_PER_BLOCK (ROWS_PER_WAVE * WAVES_PER_BLOCK)   // 128
#define BLOCK_THREADS (WAVES_PER_BLOCK * 32)               // 256

__global__ __launch_bounds__(BLOCK_THREADS)
void metak_fused_kernel(const int* __restrict__ vals,
                        const float* __restrict__ dist,
                        const float* __restrict__ w1,   // [64][32] row-major
                        const float* __restrict__ b1,   // [32]
                        const float* __restrict__ w2,   // [32][2]
                        const float* __restrict__ b2,   // [2]
                        const float* __restrict__ f1w1, // [2][4]
                        const float* __restrict__ f1b1, // [4]
                        const float* __restrict__ f1w2, // [4]
                        const float* __restrict__ f1b2, // [1]
                        float* __restrict__ out,
                        int nrows)
{
    __shared__ float sW1[64 * 32];                               // 8 KB
    __shared__ float sB1[32];
    __shared__ float sW2[32 * 2];
    __shared__ float sB2[2];
    __shared__ float sF1[17];                                    // f1w1(8) f1b1(4) f1w2(4) f1b2(1)
    __shared__ float net[WAVES_PER_BLOCK][ROWS_PER_WAVE][2 * KNN]; // 32 KB
    __shared__ float hbuf[WAVES_PER_BLOCK][ROWS_PER_WAVE][33];     // padded, ~17 KB

    const int tid  = threadIdx.x;
    const int lane = tid & 31;
    const int w    = tid >> 5;
    const int l15  = lane & 15;
    const int hi   = lane >> 4;   // 0: lanes 0-15, 1: lanes 16-31

    // ---- cooperative weight staging ----
    for (int i = tid; i < 64 * 32; i += BLOCK_THREADS) sW1[i] = w1[i];
    if (tid < 32) sB1[tid] = b1[tid];
    if (tid < 64) sW2[tid] = w2[tid];
    if (tid < 2)  sB2[tid] = b2[tid];
    if (tid < 8)       sF1[tid]  = f1w1[tid];
    else if (tid < 12) sF1[tid]  = f1b1[tid - 8];
    else if (tid < 16) sF1[tid]  = f1w2[tid - 12];
    else if (tid == 16) sF1[16]  = f1b2[0];

    // ---- Phase 1: label-count segment + feature staging (lane = K index) ----
    const int rowBase = blockIdx.x * ROWS_PER_BLOCK + w * ROWS_PER_WAVE;
    for (int r = 0; r < ROWS_PER_WAVE; ++r) {
        int grow = rowBase + r;
        int g = grow < nrows ? grow : (nrows - 1);
        int   v  = vals[(size_t)g * KNN + lane];
        float dv = dist[(size_t)g * KNN + lane];

        // is-first-occurrence of a nonzero label
        int first = (v != 0) ? 1 : 0;
        #pragma unroll
        for (int i = 0; i < KNN; ++i) {
            int vi = __shfl(v, i, 32);           // uniform broadcast -> v_readlane
            if (i < lane && vi == v) first = 0;
        }
        // inclusive prefix sum across the wave
        int cnt = first;
        #pragma unroll
        for (int off = 1; off < 32; off <<= 1) {
            int y = __shfl_up(cnt, off, 32);
            if (lane >= off) cnt += y;
        }
        net[w][r][lane]       = dv;
        net[w][r][KNN + lane] = (float)cnt;
    }
    __syncthreads();

    // ---- Phase 2: [16x64] @ [64x32] via V_WMMA_F32_16X16X4_F32, +bias, tanh ----
    #pragma unroll
    for (int nt = 0; nt < 2; ++nt) {
        v8f acc;
        float bias = sB1[nt * 16 + l15];         // bias indexed by N = l15
        #pragma unroll
        for (int p = 0; p < 8; ++p) acc[p] = bias;

        #pragma unroll
        for (int kk = 0; kk < 16; ++kk) {
            // A 16x4 layout: lanes 0-15 hold K=0,1; lanes 16-31 hold K=2,3 (M = l15)
            v2f a, bm;
            a[0]  = net[w][l15][4 * kk + 2 * hi + 0];
            a[1]  = net[w][l15][4 * kk + 2 * hi + 1];
            // B 4x16 layout mirrors A: VGPR0 -> K = 2*hi, VGPR1 -> K = 2*hi+1, N = l15
            bm[0] = sW1[(4 * kk + 2 * hi + 0) * 32 + nt * 16 + l15];
            bm[1] = sW1[(4 * kk + 2 * hi + 1) * 32 + nt * 16 + l15];
            acc = __builtin_amdgcn_wmma_f32_16x16x4_f32(
                false, a, false, bm, (short)0, acc, false, false);
        }
        // C/D layout: VGPR p -> (M = p + 8*hi, N = l15)
        #pragma unroll
        for (int p = 0; p < 8; ++p)
            hbuf[w][p + 8 * hi][nt * 16 + l15] = tanhf(acc[p]);
    }
    __syncthreads();

    // ---- Phase 3: tiny fc2->fc1 heads, one row per lane (lanes 0..15) ----
    if (lane < 16) {
        int r    = lane;
        int grow = rowBase + r;
        if (grow < nrows) {
            float o0 = sB2[0], o1 = sB2[1];
            #pragma unroll
            for (int c = 0; c < 32; ++c) {
                float hv = hbuf[w][r][c];
                o0 += hv * sW2[c * 2 + 0];
                o1 += hv * sW2[c * 2 + 1];
            }
            float o2 = sF1[16];
            #pragma unroll
            for (int q = 0; q < 4; ++q) {
                float z = tanhf(o0 * sF1[q] + o1 * sF1[4 + q] + sF1[8 + q]);
                o2 += z * sF1[12 + q];
            }
            out[(size_t)grow * 3 + 0] = o0;
            out[(size_t)grow * 3 + 1] = o1;
            out[(size_t)grow * 3 + 2] = o2;
        }
    }
}

extern "C" void kernel_launch(void* const* d_in, const int* in_sizes, int n_in,
                              void* d_out, int out_size, void* d_ws, size_t ws_size,
                              hipStream_t stream) {
    const int*   vals = (const int*)d_in[0];
    const float* dist = (const float*)d_in[1];
    const float* w1   = (const float*)d_in[2];
    const float* b1   = (const float*)d_in[3];
    const float* w2   = (const float*)d_in[4];
    const float* b2   = (const float*)d_in[5];
    const float* f1w1 = (const float*)d_in[6];
    const float* f1b1 = (const float*)d_in[7];
    const float* f1w2 = (const float*)d_in[8];
    const float* f1b2 = (const float*)d_in[9];
    float* out = (float*)d_out;

    int nrows = in_sizes[0] / KNN;                       // B*S = 32768
    int grid  = (nrows + ROWS_PER_BLOCK - 1) / ROWS_PER_BLOCK;

    hipLaunchKernelGGL(metak_fused_kernel, dim3(grid), dim3(BLOCK_THREADS), 0, stream,
                       vals, dist, w1, b1, w2, b2, f1w1, f1b1, f1w2, f1b2, out, nrows);
}